// GuidedFilter_33079838114160
// MI455X (gfx1250) — compile-verified
//
#include <hip/hip_runtime.h>

// Guided filter, R=1, EPS=0.01, fp32, NCHW 8x3x1024x1024.
// HBM-bound (~302 MB @ 23.3 TB/s ~= 13 us floor). Single fused pass per tile:
//   1) async global->LDS DMA fill of 36x36 halo tiles (ASYNCcnt path)
//   2) A,b on 34x34 from LDS, 2x2 register-blocked (4x4 window / quad)
//   3) box(A),box(b) 2x2 register-blocked, 8B nontemporal streaming stores
// 2x2 blocking halves LDS bank traffic vs naive 9-tap gathers, keeping the
// LDS roofline below the HBM roofline.

#define TW   32          // output tile width
#define TH   32          // output tile height
#define IW   36          // input tile (halo 2)
#define IH   36
#define LSTR 37          // LDS row stride (odd -> bank-parity flips per row)
#define AW   34          // A/b tile (halo 1)
#define AQ   17          // quads per dim in A/b tile (34 = 2*17)
#define EPSF 0.01f

typedef __attribute__((address_space(1))) int g_i32;   // global
typedef __attribute__((address_space(3))) int l_i32;   // LDS
typedef float v2f __attribute__((ext_vector_type(2))); // true clang vector

__device__ __forceinline__ void async_b32(float* lds_dst, const float* g_src) {
#if __has_builtin(__builtin_amdgcn_global_load_async_to_lds_b32)
    // DMA 4B/lane global->LDS, tracked by ASYNCcnt (no VGPR staging).
    __builtin_amdgcn_global_load_async_to_lds_b32(
        (g_i32*)(void*)(g_src), (l_i32*)(void*)(lds_dst), 0, 0);
#else
    *lds_dst = *g_src;
#endif
}

__device__ __forceinline__ void wait_async0() {
#if __has_builtin(__builtin_amdgcn_s_wait_asynccnt)
    __builtin_amdgcn_s_wait_asynccnt(0);
#else
    asm volatile("s_wait_asynccnt 0x0" ::: "memory");
#endif
}

// cnt in {4,6,9} for R=1 -> branchless reciprocal (no float division)
__device__ __forceinline__ float inv_cnt(int cnt) {
    float inv = 1.0f / 9.0f;
    inv = (cnt == 6) ? (1.0f / 6.0f) : inv;
    inv = (cnt == 4) ? 0.25f         : inv;
    return inv;
}

__global__ __launch_bounds__(256)
void guided_filter_kernel(const float* __restrict__ x,
                          const float* __restrict__ y,
                          float* __restrict__ out,
                          int H, int W) {
    __shared__ float sX[IH * LSTR];
    __shared__ float sY[IH * LSTR];
    __shared__ float sA[AW * LSTR];
    __shared__ float sB[AW * LSTR];

    const int tid   = threadIdx.x;
    const int x0    = blockIdx.x * TW;
    const int y0    = blockIdx.y * TH;
    const long long plane = (long long)blockIdx.z * H * W;
    const float* xp = x + plane;
    const float* yp = y + plane;

    // ---- Phase 1: fill 36x36 halo tiles of x,y (zero beyond image) ----
    for (int i = tid; i < IH * IW; i += 256) {
        const int r  = i / IW, c = i % IW;
        const int gr = y0 + r - 2, gc = x0 + c - 2;
        const int d  = r * LSTR + c;
        if ((unsigned)gr < (unsigned)H && (unsigned)gc < (unsigned)W) {
            const long long g = (long long)gr * W + gc;
            async_b32(&sX[d], xp + g);
            async_b32(&sY[d], yp + g);
        } else {
            sX[d] = 0.0f;   // zero 'SAME' padding
            sY[d] = 0.0f;
        }
    }
    wait_async0();          // ASYNCcnt -> 0 before any wave consumes LDS
    __syncthreads();

    // ---- Phase 2: A,b on 34x34, one 2x2 quad per iteration ----
    for (int q = tid; q < AQ * AQ; q += 256) {      // 289 quads
        const int qr = q / AQ, qc = q % AQ;
        const int r0 = 2 * qr, c0 = 2 * qc;         // A-grid quad origin

        // 4x4 register window of x,y (36-grid rows r0..r0+3, cols c0..c0+3)
        float xv[4][4], yv[4][4];
#pragma unroll
        for (int k = 0; k < 4; ++k)
#pragma unroll
            for (int m = 0; m < 4; ++m) {
                const int d = (r0 + k) * LSTR + (c0 + m);
                xv[k][m] = sX[d];
                yv[k][m] = sY[d];
            }

        // horizontal triple sums (shared between the two quad rows)
        float hx[4][2], hy[4][2], hxy[4][2], hxx[4][2];
#pragma unroll
        for (int k = 0; k < 4; ++k)
#pragma unroll
            for (int j = 0; j < 2; ++j) {
                hx[k][j]  = xv[k][j] + xv[k][j + 1] + xv[k][j + 2];
                hy[k][j]  = yv[k][j] + yv[k][j + 1] + yv[k][j + 2];
                hxy[k][j] = xv[k][j] * yv[k][j] + xv[k][j + 1] * yv[k][j + 1]
                          + xv[k][j + 2] * yv[k][j + 2];
                hxx[k][j] = xv[k][j] * xv[k][j] + xv[k][j + 1] * xv[k][j + 1]
                          + xv[k][j + 2] * xv[k][j + 2];
            }

#pragma unroll
        for (int i = 0; i < 2; ++i)
#pragma unroll
            for (int j = 0; j < 2; ++j) {
                const int r  = r0 + i, c = c0 + j;          // A-grid coords
                const int gr = y0 + r - 1, gc = x0 + c - 1; // image coords
                float Aval = 0.0f, Bval = 0.0f;
                if ((unsigned)gr < (unsigned)H && (unsigned)gc < (unsigned)W) {
                    const float sx  = hx[i][j]  + hx[i + 1][j]  + hx[i + 2][j];
                    const float sy  = hy[i][j]  + hy[i + 1][j]  + hy[i + 2][j];
                    const float sxy = hxy[i][j] + hxy[i + 1][j] + hxy[i + 2][j];
                    const float sxx = hxx[i][j] + hxx[i + 1][j] + hxx[i + 2][j];
                    const int nr  = 1 + (gr > 0) + (gr < H - 1);
                    const int nc  = 1 + (gc > 0) + (gc < W - 1);
                    const float inv = inv_cnt(nr * nc);     // box(q)/N == sum/cnt
                    const float mx  = sx * inv, my = sy * inv;
                    const float cov = sxy * inv - mx * my;
                    const float var = sxx * inv - mx * mx;
                    Aval = cov / (var + EPSF);
                    Bval = my - Aval * mx;
                }
                sA[r * LSTR + c] = Aval;
                sB[r * LSTR + c] = Bval;
            }
    }
    __syncthreads();

    // ---- Phase 3: box(A), box(b), combine; exactly one 2x2 quad/thread ----
    {
        const int qr = tid >> 4, qc = tid & 15;
        const int r0 = 2 * qr, c0 = 2 * qc;          // output-tile quad origin

        float Av[4][4], Bv[4][4];
#pragma unroll
        for (int k = 0; k < 4; ++k)
#pragma unroll
            for (int m = 0; m < 4; ++m) {
                const int d = (r0 + k) * LSTR + (c0 + m);  // A-grid window
                Av[k][m] = sA[d];
                Bv[k][m] = sB[d];
            }

        float hA[4][2], hB[4][2];
#pragma unroll
        for (int k = 0; k < 4; ++k)
#pragma unroll
            for (int j = 0; j < 2; ++j) {
                hA[k][j] = Av[k][j] + Av[k][j + 1] + Av[k][j + 2];
                hB[k][j] = Bv[k][j] + Bv[k][j + 1] + Bv[k][j + 2];
            }

        float o[2][2];
#pragma unroll
        for (int i = 0; i < 2; ++i)
#pragma unroll
            for (int j = 0; j < 2; ++j) {
                const int r  = r0 + i, c = c0 + j;    // tile coords (in-image)
                const int gr = y0 + r, gc = x0 + c;
                const float sa = hA[i][j] + hA[i + 1][j] + hA[i + 2][j];
                const float sb = hB[i][j] + hB[i + 1][j] + hB[i + 2][j];
                const int nr  = 1 + (gr > 0) + (gr < H - 1);
                const int nc  = 1 + (gc > 0) + (gc < W - 1);
                const float inv = inv_cnt(nr * nc);
                const float xc  = sX[(r + 2) * LSTR + (c + 2)];
                o[i][j] = (sa * inv) * xc + (sb * inv);
            }

        // packed 8B NT stores: contiguous 128B per half-wave, global_store_b64
        const long long g0 = plane + (long long)(y0 + r0) * W + (x0 + c0);
        v2f v0; v0.x = o[0][0]; v0.y = o[0][1];
        v2f v1; v1.x = o[1][0]; v1.y = o[1][1];
        __builtin_nontemporal_store(v0, (v2f*)(out + g0));
        __builtin_nontemporal_store(v1, (v2f*)(out + g0 + W));
    }
}

extern "C" void kernel_launch(void* const* d_in, const int* in_sizes, int n_in,
                              void* d_out, int out_size, void* d_ws, size_t ws_size,
                              hipStream_t stream) {
    const float* x = (const float*)d_in[0];
    const float* y = (const float*)d_in[1];
    float* out     = (float*)d_out;
    const int H = 1024, W = 1024;
    const int planes = in_sizes[0] / (H * W);   // 8 * 3 = 24
    dim3 grid(W / TW, H / TH, planes);
    guided_filter_kernel<<<grid, dim3(256), 0, stream>>>(x, y, out, H, W);
}